// DepthTracker_67087389163943
// MI455X (gfx1250) — compile-verified
//
#include <hip/hip_runtime.h>
#include <hip/hip_bf16.h>

typedef float v2f __attribute__((ext_vector_type(2)));
typedef float v8f __attribute__((ext_vector_type(8)));

#define S2   49      // 7x7 support
#define CF   128     // channels
#define TS   52      // T/F LDS stride (49 data + col49 zero pad; 2*52%64=40 -> conflict-free)
#define VK   2432    // padded correlation-row length (2401 -> 16*152)
#define XS   20      // X chunk LDS stride (16 + 4)
#define HS   388     // hidden LDS stride (384 + 4; 388%64=4 -> conflict-free A loads)
#define WS1  392     // w1 chunk LDS stride (2*392%64=16 -> conflict-free)
#define W2S  264     // w2 chunk LDS stride (2*264%64=16 -> conflict-free)

#define STAGE_OFF 24832          // Hl floats (64*388)
#define ABUF      7552           // per-buffer floats phase A: Xc(64*20) + Wc(16*392)
#define BBUF      8448           // per-buffer floats phase B: W2c(32*264)
#define SMEM_FLOATS (STAGE_OFF + 2 * BBUF)   // 41728 floats = 166,912 B (CDNA5: 320KB/WGP)

#ifndef __has_builtin
#define __has_builtin(x) 0
#endif
#if __has_builtin(__builtin_amdgcn_global_load_async_to_lds_b128)
#define USE_ASYNC 1
#else
#define USE_ASYNC 0
#endif

#define AS1 __attribute__((address_space(1)))
#define AS3 __attribute__((address_space(3)))

// exact pointee type of the async builtin params (per hipcc diagnostic):
// int __attribute__((vector_size(4*sizeof(int))))
typedef int a4i __attribute__((vector_size(4 * sizeof(int))));

// 16-byte copy global -> LDS (async direct-to-LDS when available)
__device__ __forceinline__ void copy16_to_lds(float* lds, const float* g) {
#if USE_ASYNC
    __builtin_amdgcn_global_load_async_to_lds_b128(
        (AS1 a4i*)g, (AS3 a4i*)lds, 0, 0);
#else
    const float4 v = *(const float4*)g;
    lds[0] = v.x; lds[1] = v.y; lds[2] = v.z; lds[3] = v.w;
#endif
}

__device__ __forceinline__ void wait_async_lds() {
#if USE_ASYNC
#if __has_builtin(__builtin_amdgcn_s_wait_asynccnt)
    __builtin_amdgcn_s_wait_asynccnt(0);
#else
    asm volatile("s_wait_asynccnt 0x0" ::: "memory");
#endif
#endif
}

// ---------------------------------------------------------------------------
// Kernel 1: per (track, level) -> bilinear sample + 49x49 correlation (WMMA)
// ---------------------------------------------------------------------------
__global__ __launch_bounds__(256) void dt_corr_kernel(
    const float* __restrict__ coords,
    const float* __restrict__ fmap0,
    const float* __restrict__ fmap1,
    const float* __restrict__ fmap2,
    const float* __restrict__ fmap3,
    const float* __restrict__ tfeat,
    float* __restrict__ vol)
{
    __shared__ float Tl[CF * TS];      // T in [c][h] layout, col 49..51 zeroed
    __shared__ float Fl[CF * TS];      // F in [c][p] layout, col 49..51 zeroed
    __shared__ int   pidx[S2][4];
    __shared__ float pw[S2][4];

    const int wg  = blockIdx.x;
    const int n   = wg >> 2;
    const int lvl = wg & 3;
    const int tid = threadIdx.x;

    const int W  = 128 >> lvl;
    const int H  = 96  >> lvl;
    const int HW = H * W;
    const float* fmap = (lvl == 0) ? fmap0 : (lvl == 1) ? fmap1 : (lvl == 2) ? fmap2 : fmap3;

    const float scale = 1.0f / (float)(4 << lvl);
    const float cx = coords[2 * n + 0] * scale;
    const float cy = coords[2 * n + 1] * scale;

    // Per-support-point bilinear indices/weights (floor before clamp, weights
    // from unclamped floor -- matches reference exactly).
    if (tid < S2) {
        const int p = tid;
        const float dx = (float)(p / 7) - 3.0f;
        const float dy = (float)(p % 7) - 3.0f;
        const float x = cx + dx, y = cy + dy;
        const float x0f = floorf(x), y0f = floorf(y);
        const float wx = x - x0f,   wy = y - y0f;
        int x0 = (int)x0f; x0 = x0 < 0 ? 0 : (x0 > W - 1 ? W - 1 : x0);
        int x1 = (x0 + 1 > W - 1) ? W - 1 : x0 + 1;
        int y0 = (int)y0f; y0 = y0 < 0 ? 0 : (y0 > H - 1 ? H - 1 : y0);
        int y1 = (y0 + 1 > H - 1) ? H - 1 : y0 + 1;
        pidx[p][0] = y0 * W + x0;
        pidx[p][1] = y0 * W + x1;
        pidx[p][2] = y1 * W + x0;
        pidx[p][3] = y1 * W + x1;
        pw[p][0] = (1.0f - wx) * (1.0f - wy);
        pw[p][1] = wx * (1.0f - wy);
        pw[p][2] = (1.0f - wx) * wy;
        pw[p][3] = wx * wy;
    }

    // Zero padding columns 49..51 so clamped indices read 0.0f (no guards in hot loop).
    for (int e = tid; e < CF * 3; e += 256) {
        const int c = e / 3;
        const int j = e - c * 3;
        Tl[c * TS + S2 + j] = 0.0f;
        Fl[c * TS + S2 + j] = 0.0f;
    }

    // Zero the padded tail of this output row (cols 2401..2431).
    if (tid < VK - 2401)
        vol[(size_t)wg * VK + 2401 + tid] = 0.0f;

    // Load T = track_feats[lvl][n] : contiguous [128][49].
    const float* tf = tfeat + ((size_t)(lvl * 2048 + n)) * (CF * S2);
    for (int e = tid; e < CF * S2; e += 256) {
        const int c = e / S2;
        const int h = e - c * S2;
        Tl[c * TS + h] = tf[e];
    }
    __syncthreads();

    // Bilinear-sample F[c][p].
    for (int e = tid; e < CF * S2; e += 256) {
        const int c = e / S2;
        const int p = e - c * S2;
        const float* base = fmap + (size_t)c * HW;
        Fl[c * TS + p] = pw[p][0] * base[pidx[p][0]] + pw[p][1] * base[pidx[p][1]]
                       + pw[p][2] * base[pidx[p][2]] + pw[p][3] * base[pidx[p][3]];
    }
    __syncthreads();

    // vol = T^T F : 49x49 padded to 64x64 = 16 tiles; wave w owns m-tile (w>>1)
    // and n-tiles {2*(w&1), 2*(w&1)+1} -> one shared A frag, 2 B frags, 2 WMMAs.
    const int wave = tid >> 5;
    const int lane = tid & 31;
    const int half = lane >> 4;
    const int lm   = lane & 15;

    const int tm  = wave >> 1;
    const int tn0 = (wave & 1) * 2;
    const int m   = tm * 16 + lm;
    const int mc  = (m  < S2) ? m  : S2;     // clamp to zero column
    const int n0  = tn0 * 16 + lm;
    const int n0c = (n0 < S2) ? n0 : S2;
    const int n1  = n0 + 16;
    const int n1c = (n1 < S2) ? n1 : S2;

    v8f acc0 = {}, acc1 = {};
#pragma unroll 8
    for (int kk = 0; kk < CF; kk += 4) {
        const int r0i = (kk + 2 * half) * TS;
        const int r1i = r0i + TS;
        v2f a, b0, b1;
        a.x  = Tl[r0i + mc];  a.y  = Tl[r1i + mc];
        b0.x = Fl[r0i + n0c]; b0.y = Fl[r1i + n0c];
        b1.x = Fl[r0i + n1c]; b1.y = Fl[r1i + n1c];
        acc0 = __builtin_amdgcn_wmma_f32_16x16x4_f32(
            false, a, false, b0, (short)0, acc0, false, false);
        acc1 = __builtin_amdgcn_wmma_f32_16x16x4_f32(
            false, a, false, b1, (short)0, acc1, false, false);
    }

    float* vrow = vol + (size_t)wg * VK;
    for (int v = 0; v < 8; ++v) {
        const int mg = tm * 16 + v + 8 * half;
        if (mg < S2) {
            if (n0 < S2) vrow[mg * S2 + n0] = acc0[v];
            if (n1 < S2) vrow[mg * S2 + n1] = acc1[v];
        }
    }
}

// ---------------------------------------------------------------------------
// Kernel 2 staging helpers (async direct-to-LDS, double buffered)
// ---------------------------------------------------------------------------
__device__ __forceinline__ void stageA(float* Xc, float* Wc,
                                       const float* __restrict__ vol,
                                       const float* __restrict__ w1,
                                       int r0, int kc, int tid)
{
    // X chunk 64x16: one 16B copy per thread
    {
        const int r = tid >> 2;
        const int q = (tid & 3) * 4;
        copy16_to_lds(&Xc[r * XS + q], vol + (size_t)(r0 + r) * VK + kc + q);
    }
    // w1 chunk 16x384: six 16B copies per thread, guard k >= 2401
    {
        const int kr = tid >> 4;
        const int cb = (tid & 15) * 24;
        const int kg = kc + kr;
        if (kg < 2401) {
            for (int i = 0; i < 6; ++i)
                copy16_to_lds(&Wc[kr * WS1 + cb + i * 4],
                              w1 + (size_t)kg * 384 + cb + i * 4);
        } else {
            for (int i = 0; i < 24; ++i) Wc[kr * WS1 + cb + i] = 0.0f;
        }
    }
}

__device__ __forceinline__ void stageB(float* W2c,
                                       const float* __restrict__ w2,
                                       int kc, int tid)
{
    // w2 chunk 32x256: eight 16B copies per thread
    const int kr = tid >> 3;
    const int cb = (tid & 7) * 32;
    for (int i = 0; i < 8; ++i)
        copy16_to_lds(&W2c[kr * W2S + cb + i * 4],
                      w2 + (size_t)(kc + kr) * 256 + cb + i * 4);
}

// ---------------------------------------------------------------------------
// Kernel 2: fused MLP over all 8192 rows:
//   O = gelu(X(8192x2401) @ w1 + b1) @ w2 + b2, 64 rows per workgroup (WMMA)
// ---------------------------------------------------------------------------
__global__ __launch_bounds__(256) void dt_mlp_kernel(
    const float* __restrict__ vol,
    const float* __restrict__ w1,
    const float* __restrict__ b1,
    const float* __restrict__ w2,
    const float* __restrict__ b2,
    float* __restrict__ out)
{
    extern __shared__ float smem[];
    float* Hl = smem;                                   // 64*388 floats

    const int wg  = blockIdx.x;
    const int r0  = wg * 64;
    const int tid = threadIdx.x;
    const int wave = tid >> 5;
    const int lane = tid & 31;
    const int half = lane >> 4;
    const int lm   = lane & 15;

    // ---------------- Phase A: H = gelu(X @ w1 + b1) ----------------
    v8f accA[3][4] = {};

    // double-buffered staging: buffer b at STAGE_OFF + b*ABUF (Xc then Wc)
    stageA(smem + STAGE_OFF, smem + STAGE_OFF + 1280, vol, w1, r0, 0, tid);
    wait_async_lds();
    __syncthreads();

    for (int ci = 0; ci < VK / 16; ++ci) {
        const int kc  = ci * 16;
        const int cur = ci & 1;
        float* Xc = smem + STAGE_OFF + cur * ABUF;
        float* Wc = Xc + 1280;

        if (ci + 1 < VK / 16) {
            float* Xn = smem + STAGE_OFF + (cur ^ 1) * ABUF;
            stageA(Xn, Xn + 1280, vol, w1, r0, kc + 16, tid);
        }

#pragma unroll
        for (int k4 = 0; k4 < 16; k4 += 4) {
            v2f a[4], b[3];
            for (int mt = 0; mt < 4; ++mt) {
                const int base = (mt * 16 + lm) * XS + k4 + 2 * half;
                a[mt].x = Xc[base];
                a[mt].y = Xc[base + 1];
            }
            for (int j = 0; j < 3; ++j) {
                const int nc = (wave + j * 8) * 16 + lm;
                b[j].x = Wc[(k4 + 2 * half)     * WS1 + nc];
                b[j].y = Wc[(k4 + 2 * half + 1) * WS1 + nc];
            }
            for (int j = 0; j < 3; ++j)
                for (int mt = 0; mt < 4; ++mt)
                    accA[j][mt] = __builtin_amdgcn_wmma_f32_16x16x4_f32(
                        false, a[mt], false, b[j], (short)0, accA[j][mt], false, false);
        }

        wait_async_lds();   // next chunk landed (hidden behind the WMMAs above)
        __syncthreads();
    }

    // Prefetch phase-B chunk 0 so the gelu epilogue hides its latency.
    stageB(smem + STAGE_OFF, w2, 0, tid);

    // bias + tanh-gelu -> Hl
    for (int j = 0; j < 3; ++j) {
        const int col = (wave + j * 8) * 16 + lm;
        const float bias = b1[col];
        for (int mt = 0; mt < 4; ++mt) {
            for (int v = 0; v < 8; ++v) {
                const int m = mt * 16 + v + 8 * half;
                const float x = accA[j][mt][v] + bias;
                const float g = 0.5f * x *
                    (1.0f + tanhf(0.7978845608028654f * (x + 0.044715f * x * x * x)));
                Hl[m * HS + col] = g;
            }
        }
    }
    wait_async_lds();
    __syncthreads();

    // ---------------- Phase B: O = H @ w2 + b2 ----------------
    v8f accB[2][4] = {};

    for (int ci = 0; ci < 12; ++ci) {
        const int kc  = ci * 32;
        const int cur = ci & 1;
        float* W2c = smem + STAGE_OFF + cur * BBUF;

        if (ci + 1 < 12)
            stageB(smem + STAGE_OFF + (cur ^ 1) * BBUF, w2, kc + 32, tid);

#pragma unroll
        for (int k4 = 0; k4 < 32; k4 += 4) {
            v2f a[4], b[2];
            for (int mt = 0; mt < 4; ++mt) {
                const int base = (mt * 16 + lm) * HS + kc + k4 + 2 * half;
                a[mt].x = Hl[base];
                a[mt].y = Hl[base + 1];
            }
            for (int j = 0; j < 2; ++j) {
                const int nc = (wave + j * 8) * 16 + lm;
                b[j].x = W2c[(k4 + 2 * half)     * W2S + nc];
                b[j].y = W2c[(k4 + 2 * half + 1) * W2S + nc];
            }
            for (int j = 0; j < 2; ++j)
                for (int mt = 0; mt < 4; ++mt)
                    accB[j][mt] = __builtin_amdgcn_wmma_f32_16x16x4_f32(
                        false, a[mt], false, b[j], (short)0, accB[j][mt], false, false);
        }

        wait_async_lds();
        __syncthreads();
    }

    // bias + scatter to output: row R = n*4+lvl -> out[n*1024 + lvl*256 + col]
    for (int j = 0; j < 2; ++j) {
        const int col = (wave + j * 8) * 16 + lm;
        const float bias = b2[col];
        for (int mt = 0; mt < 4; ++mt) {
            for (int v = 0; v < 8; ++v) {
                const int m = mt * 16 + v + 8 * half;
                const int R = r0 + m;
                const int trk = R >> 2;
                const int lv  = R & 3;
                out[(size_t)trk * 1024 + lv * 256 + col] = accB[j][mt][v] + bias;
            }
        }
    }
}

// ---------------------------------------------------------------------------
extern "C" void kernel_launch(void* const* d_in, const int* in_sizes, int n_in,
                              void* d_out, int out_size, void* d_ws, size_t ws_size,
                              hipStream_t stream) {
    const float* coords = (const float*)d_in[0];
    const float* fmap0  = (const float*)d_in[1];
    const float* fmap1  = (const float*)d_in[2];
    const float* fmap2  = (const float*)d_in[3];
    const float* fmap3  = (const float*)d_in[4];
    const float* tfeat  = (const float*)d_in[5];
    const float* w1     = (const float*)d_in[6];
    const float* b1     = (const float*)d_in[7];
    const float* w2     = (const float*)d_in[8];
    const float* b2     = (const float*)d_in[9];
    float* vol = (float*)d_ws;          // 8192 x 2432 fp32 = ~80 MB
    float* out = (float*)d_out;         // 2048 x 1024 fp32

    dt_corr_kernel<<<8192, 256, 0, stream>>>(coords, fmap0, fmap1, fmap2, fmap3,
                                             tfeat, vol);

    const size_t smem_bytes = (size_t)SMEM_FLOATS * sizeof(float); // ~163 KB
    dt_mlp_kernel<<<128, 256, smem_bytes, stream>>>(vol, w1, b1, w2, b2, out);
}